// _DotAttention_12094627906171
// MI455X (gfx1250) — compile-verified
//
#include <hip/hip_runtime.h>
#include <hip/hip_bf16.h>
#include <cstddef>

// ---------------------------------------------------------------------------
// Single-query attention, algebraically refactored:
//   logits[b,m] = tokens[b,m] . ((query@q_w.T + q_b) @ k_w) * SCALE  (+const -> cancels)
//   out[b]     = (sum_m softmax_m * tokens[b,m]) @ v_w.T + v_b
// Memory-bound: one 268 MB streaming pass (~11.5us at 23.3 TB/s).
// Small 32x256x256 projection GEMMs use V_WMMA_F32_16X16X4_F32 (exact fp32).
// Token tiles staged into LDS with GLOBAL_LOAD_ASYNC_TO_LDS_B128 (ASYNCcnt)
// when the builtin is available; otherwise VGPR->ds_store fallback.
// ---------------------------------------------------------------------------

#define D_MODEL 256
#define SCALE   0.0625f          // 1/sqrt(256)
#define CHUNK   64               // tokens per stream block
#define NCHUNK  128              // 8192 / 64
#define NBATCH  32

#if defined(__has_builtin)
#  if __has_builtin(__builtin_amdgcn_global_load_async_to_lds_b128) && \
      __has_builtin(__builtin_amdgcn_s_wait_asynccnt)
#    define USE_ASYNC_LDS 1
#  endif
#endif

typedef float v2f __attribute__((ext_vector_type(2)));
typedef float v8f __attribute__((ext_vector_type(8)));
typedef int   v4i __attribute__((ext_vector_type(4)));

#define AS1 __attribute__((address_space(1)))
#define AS3 __attribute__((address_space(3)))

__device__ __forceinline__ float wave_sum(float v) {
    #pragma unroll
    for (int m = 16; m; m >>= 1) v += __shfl_xor(v, m, 32);
    return v;
}
__device__ __forceinline__ float wave_max(float v) {
    #pragma unroll
    for (int m = 16; m; m >>= 1) v = fmaxf(v, __shfl_xor(v, m, 32));
    return v;
}

__device__ __forceinline__ v8f wmma_f32(v2f a, v2f b, v8f c) {
    // D = A(16x4) * B(4x16) + C(16x16), fp32
    return __builtin_amdgcn_wmma_f32_16x16x4_f32(
        false, a, false, b, (short)0, c, false, false);
}

// ---------------------------------------------------------------------------
// Kernel 1: Q = query @ q_w.T + q_b ;  qks = (Q @ k_w) * SCALE
// One block, 32 waves; wave w owns output tile (m0 = 16*(w/16), n0 = 16*(w%16)).
// ---------------------------------------------------------------------------
__global__ __launch_bounds__(1024) void prep_kernel(
    const float* __restrict__ query, const float* __restrict__ q_w,
    const float* __restrict__ q_b,   const float* __restrict__ k_w,
    float* __restrict__ qks)
{
    __shared__ alignas(16) float sQ[NBATCH * D_MODEL];   // 32 KB
    const int tid  = threadIdx.x;
    const int lane = tid & 31, wave = tid >> 5;
    const int r    = lane & 15, half = lane >> 4;
    const int m0   = (wave >> 4) * 16;       // 0 or 16 (batch rows)
    const int n0   = (wave & 15) * 16;       // output dim tile

    // --- GEMM 1: Q[m,n] = sum_k query[m,k] * q_w[n,k]  (B[k,n] = q_w[n,k]) ---
    v8f c = {0.f,0.f,0.f,0.f,0.f,0.f,0.f,0.f};
    for (int k = 0; k < D_MODEL; k += 4) {
        v2f a = *(const v2f*)(query + (m0 + r) * D_MODEL + k + 2 * half);
        v2f b = *(const v2f*)(q_w   + (n0 + r) * D_MODEL + k + 2 * half);
        c = wmma_f32(a, b, c);
    }
    const float qb = q_b[n0 + r];
    const int rowBase = m0 + half * 8;
    #pragma unroll
    for (int i = 0; i < 8; ++i)
        sQ[(rowBase + i) * D_MODEL + n0 + r] = c[i] + qb;
    __syncthreads();

    // --- GEMM 2: qks[m,n] = SCALE * sum_k Q[m,k] * k_w[k,n]  (B[k,n]=k_w[k*256+n]) ---
    v8f c2 = {0.f,0.f,0.f,0.f,0.f,0.f,0.f,0.f};
    for (int k = 0; k < D_MODEL; k += 4) {
        v2f a = *(const v2f*)(&sQ[(m0 + r) * D_MODEL + k + 2 * half]);
        v2f b;
        b.x = k_w[(k + 2 * half)     * D_MODEL + n0 + r];
        b.y = k_w[(k + 2 * half + 1) * D_MODEL + n0 + r];
        c2 = wmma_f32(a, b, c2);
    }
    #pragma unroll
    for (int i = 0; i < 8; ++i)
        qks[(rowBase + i) * D_MODEL + n0 + r] = c2[i] * SCALE;
}

// ---------------------------------------------------------------------------
// Kernel 2: streaming split-K flash pass. Block = (chunk, batch), 256 threads.
// Emits per-chunk partial (max, sum, acc[256] = sum exp(l-max)*token).
// ---------------------------------------------------------------------------
__global__ __launch_bounds__(256) void stream_kernel(
    const float* __restrict__ tokens, const float* __restrict__ qks,
    float* __restrict__ pmax, float* __restrict__ psum, float* __restrict__ pacc)
{
    __shared__ alignas(16) float sQK[D_MODEL];
    __shared__ alignas(16) float sT[CHUNK * D_MODEL];    // 64 KB token tile
    __shared__ float sLog[CHUNK];
    __shared__ float sExp[CHUNK];

    const int tid  = threadIdx.x;
    const int lane = tid & 31, wave = tid >> 5;
    const int chunk = blockIdx.x, b = blockIdx.y;

    sQK[tid] = qks[b * D_MODEL + tid];

    const size_t base = ((size_t)b * 8192 + (size_t)chunk * CHUNK) * D_MODEL;

#if defined(USE_ASYNC_LDS)
    // --- Stage the 64x256 fp32 tile straight into LDS (no VGPR round-trip),
    //     tracked by ASYNCcnt. 16 x b128 per thread = whole 64 KB tile. ---
    {
        const float* gsrc = tokens + base;
        #pragma unroll
        for (int j = 0; j < 16; ++j) {
            const int f4 = tid + j * 256;            // float4 index in tile
            __builtin_amdgcn_global_load_async_to_lds_b128(
                (AS1 v4i*)(gsrc + (size_t)f4 * 4),
                (AS3 v4i*)(&sT[f4 * 4]),
                0, 0);
        }
        __builtin_amdgcn_s_wait_asynccnt(0);
    }
    __syncthreads();   // covers sQK and all waves' staged sT

    const float4* qk4 = (const float4*)sQK;
    const float4 q0 = qk4[lane], q1 = qk4[lane + 32];
    const float4* tile4 = (const float4*)sT;

    // Phase 1: logits for 8 tokens per wave, reading the staged LDS tile.
    #pragma unroll
    for (int i = 0; i < 8; ++i) {
        const int m = wave * 8 + i;
        float4 x0 = tile4[m * 64 + lane];
        float4 x1 = tile4[m * 64 + lane + 32];
        float d = x0.x*q0.x + x0.y*q0.y + x0.z*q0.z + x0.w*q0.w
                + x1.x*q1.x + x1.y*q1.y + x1.z*q1.z + x1.w*q1.w;
        d = wave_sum(d);
        if (lane == 0) sLog[m] = d;
    }
#else
    __syncthreads();

    const float4* qk4 = (const float4*)sQK;
    const float4 q0 = qk4[lane], q1 = qk4[lane + 32];

    // Phase 1: logits for 8 tokens per wave; stage tokens in LDS via VGPRs.
    #pragma unroll
    for (int i = 0; i < 8; ++i) {
        const int m = wave * 8 + i;
        const float4* row = (const float4*)(tokens + base + (size_t)m * D_MODEL);
        float4 x0 = row[lane], x1 = row[lane + 32];
        ((float4*)(sT + m * D_MODEL))[lane]      = x0;
        ((float4*)(sT + m * D_MODEL))[lane + 32] = x1;
        float d = x0.x*q0.x + x0.y*q0.y + x0.z*q0.z + x0.w*q0.w
                + x1.x*q1.x + x1.y*q1.y + x1.z*q1.z + x1.w*q1.w;
        d = wave_sum(d);
        if (lane == 0) sLog[m] = d;
    }
#endif
    __syncthreads();

    // Phase 2: local softmax stats (wave 0).
    if (wave == 0) {
        float l0 = sLog[lane], l1 = sLog[lane + 32];
        float mx = wave_max(fmaxf(l0, l1));
        float e0 = __expf(l0 - mx), e1 = __expf(l1 - mx);
        sExp[lane] = e0; sExp[lane + 32] = e1;
        float s = wave_sum(e0 + e1);
        if (lane == 0) {
            pmax[b * NCHUNK + chunk] = mx;
            psum[b * NCHUNK + chunk] = s;
        }
    }
    __syncthreads();

    // Phase 3: acc[t] = sum_m exp_m * token[m][t]  (conflict-free LDS reads).
    float acc = 0.f;
    #pragma unroll 8
    for (int m = 0; m < CHUNK; ++m)
        acc += sExp[m] * sT[m * D_MODEL + tid];
    pacc[(size_t)(b * NCHUNK + chunk) * D_MODEL + tid] = acc;
}

// ---------------------------------------------------------------------------
// Kernel 3: merge 128 partials per batch -> normalized weighted token average.
// ---------------------------------------------------------------------------
__global__ __launch_bounds__(256) void reduce_kernel(
    const float* __restrict__ pmax, const float* __restrict__ psum,
    const float* __restrict__ pacc, float* __restrict__ tavg)
{
    __shared__ float sPM[NCHUNK], sPS[NCHUNK], sCoef[NCHUNK];
    const int tid = threadIdx.x, lane = tid & 31, wave = tid >> 5;
    const int b = blockIdx.x;

    if (tid < NCHUNK) {
        sPM[tid] = pmax[b * NCHUNK + tid];
        sPS[tid] = psum[b * NCHUNK + tid];
    }
    __syncthreads();
    if (wave == 0) {
        float m0 = sPM[lane], m1 = sPM[lane+32], m2 = sPM[lane+64], m3 = sPM[lane+96];
        float mx = wave_max(fmaxf(fmaxf(m0, m1), fmaxf(m2, m3)));
        float s = sPS[lane]    * __expf(m0 - mx) + sPS[lane+32] * __expf(m1 - mx)
                + sPS[lane+64] * __expf(m2 - mx) + sPS[lane+96] * __expf(m3 - mx);
        s = wave_sum(s);
        const float inv = 1.f / s;
        sCoef[lane]      = __expf(m0 - mx) * inv;
        sCoef[lane + 32] = __expf(m1 - mx) * inv;
        sCoef[lane + 64] = __expf(m2 - mx) * inv;
        sCoef[lane + 96] = __expf(m3 - mx) * inv;
    }
    __syncthreads();

    float acc = 0.f;
    const float* pb = pacc + (size_t)b * NCHUNK * D_MODEL;
    for (int c = 0; c < NCHUNK; ++c)
        acc += sCoef[c] * pb[(size_t)c * D_MODEL + tid];
    tavg[b * D_MODEL + tid] = acc;
}

// ---------------------------------------------------------------------------
// Kernel 4: out = Tavg @ v_w.T + v_b via WMMA f32 (B[k,n] = v_w[n,k]).
// ---------------------------------------------------------------------------
__global__ __launch_bounds__(1024) void out_kernel(
    const float* __restrict__ tavg, const float* __restrict__ v_w,
    const float* __restrict__ v_b, float* __restrict__ out)
{
    __shared__ alignas(16) float sT[NBATCH * D_MODEL];
    const int tid = threadIdx.x;
    #pragma unroll
    for (int i = 0; i < 8; ++i) sT[tid + i * 1024] = tavg[tid + i * 1024];
    __syncthreads();

    const int lane = tid & 31, wave = tid >> 5;
    const int r = lane & 15, half = lane >> 4;
    const int m0 = (wave >> 4) * 16, n0 = (wave & 15) * 16;

    v8f c = {0.f,0.f,0.f,0.f,0.f,0.f,0.f,0.f};
    for (int k = 0; k < D_MODEL; k += 4) {
        v2f a = *(const v2f*)(&sT[(m0 + r) * D_MODEL + k + 2 * half]);
        v2f b = *(const v2f*)(v_w + (n0 + r) * D_MODEL + k + 2 * half);
        c = wmma_f32(a, b, c);
    }
    const float vb = v_b[n0 + r];
    const int rowBase = m0 + half * 8;
    #pragma unroll
    for (int i = 0; i < 8; ++i)
        out[(rowBase + i) * D_MODEL + n0 + r] = c[i] + vb;
}

// ---------------------------------------------------------------------------
extern "C" void kernel_launch(void* const* d_in, const int* in_sizes, int n_in,
                              void* d_out, int out_size, void* d_ws, size_t ws_size,
                              hipStream_t stream) {
    const float* query  = (const float*)d_in[0];
    const float* tokens = (const float*)d_in[1];
    const float* q_w    = (const float*)d_in[2];
    const float* q_b    = (const float*)d_in[3];
    const float* k_w    = (const float*)d_in[4];
    // d_in[5] = k_b: constant logit shift per batch -> cancels in softmax.
    const float* v_w    = (const float*)d_in[6];
    const float* v_b    = (const float*)d_in[7];

    float* ws   = (float*)d_ws;
    float* qks  = ws;                                    //  8192
    float* pmax = qks  + NBATCH * D_MODEL;               //  4096
    float* psum = pmax + NBATCH * NCHUNK;                //  4096
    float* pacc = psum + NBATCH * NCHUNK;                //  1048576
    float* tavg = pacc + (size_t)NBATCH * NCHUNK * D_MODEL; // 8192
    // total ~4.2 MB of workspace

    prep_kernel  <<<1, 1024, 0, stream>>>(query, q_w, q_b, k_w, qks);
    stream_kernel<<<dim3(NCHUNK, NBATCH), 256, 0, stream>>>(tokens, qks, pmax, psum, pacc);
    reduce_kernel<<<NBATCH, 256, 0, stream>>>(pmax, psum, pacc, tavg);
    out_kernel   <<<1, 1024, 0, stream>>>(tavg, v_w, v_b, (float*)d_out);
}